// LogCentroidModule_6356551598191
// MI455X (gfx1250) — compile-verified
//
#include <hip/hip_runtime.h>
#include <hip/hip_bf16.h>

typedef float v2f __attribute__((ext_vector_type(2)));
typedef float v8f __attribute__((ext_vector_type(8)));

#define NB 8192
#define NU 512
#define NF 64
#define NTOPK 8

// ---------------- Kernel 0: row norms of x and c -> workspace ----------------
__global__ void norms_kernel(const float* __restrict__ x,
                             const float* __restrict__ c,
                             float* __restrict__ xnorm,
                             float* __restrict__ cnorm) {
    int i = blockIdx.x * blockDim.x + threadIdx.x;
    const float* src;
    float* dst;
    if (i < NB) {
        src = x + (size_t)i * NF;
        dst = xnorm + i;
    } else if (i < NB + NU) {
        src = c + (size_t)(i - NB) * NF;
        dst = cnorm + (i - NB);
    } else {
        return;
    }
    float s = 0.0f;
    #pragma unroll
    for (int k = 0; k < NF; k += 4) {
        float4 v = *(const float4*)(src + k);
        s += v.x * v.x + v.y * v.y + v.z * v.z + v.w * v.w;
    }
    *dst = s;
}

// ---------------- Kernel 1: d = sqrt(|x|^2 + |c|^2 - 2 x.c) via WMMA ----------------
// One wave (32 lanes) computes one 16x16 tile of d.
// A (16x4 f32) layout: lanes 0-15 M=lane, VGPR0=K0,VGPR1=K1; lanes 16-31 M=lane-16, K2/K3.
// B (4x16 f32) layout mirrors it: VGPR0 = row K=2*half, VGPR1 = row K=2*half+1, N=lane&15.
// C/D (16x16 f32): VGPR r -> M = r (lanes 0-15) or M = r+8 (lanes 16-31), N = lane&15.
__global__ void dist_wmma_kernel(const float* __restrict__ x,
                                 const float* __restrict__ c,
                                 const float* __restrict__ xnorm,
                                 const float* __restrict__ cnorm,
                                 float* __restrict__ d) {
    const int waveId = threadIdx.x >> 5;
    const int lane   = threadIdx.x & 31;
    const int tile   = blockIdx.x * (blockDim.x >> 5) + waveId;
    const int tile_m = tile >> 5;   // 512 tiles along B
    const int tile_n = tile & 31;   // 32 tiles along N_UNITS

    const int half = lane >> 4;     // 0: K pair {0,1}, 1: K pair {2,3}
    const int l    = lane & 15;

    const float* xrow = x + (size_t)(tile_m * 16 + l) * NF; // A: M = l
    const float* crow = c + (size_t)(tile_n * 16 + l) * NF; // B: N = l

    v8f acc = {};
    #pragma unroll
    for (int kk = 0; kk < NF; kk += 4) {
        v2f a = *(const v2f*)(xrow + kk + 2 * half);
        v2f b = *(const v2f*)(crow + kk + 2 * half);
        acc = __builtin_amdgcn_wmma_f32_16x16x4_f32(
            /*neg_a=*/false, a, /*neg_b=*/false, b,
            /*c_mod=*/(short)0, acc, /*reuse_a=*/false, /*reuse_b=*/false);
    }

    const float cn = cnorm[tile_n * 16 + l];     // this lane's column norm
    const int n_global = tile_n * 16 + l;

    #pragma unroll
    for (int r = 0; r < 8; ++r) {
        const int m = r + half * 8;              // C layout: VGPR r -> M
        const int m_global = tile_m * 16 + m;
        const float xn = xnorm[m_global];
        float d2 = xn + cn - 2.0f * acc[r];
        d2 = fmaxf(d2, 0.0f);
        d[(size_t)m_global * NU + n_global] = sqrtf(d2);
    }
}

// ---------------- Kernel 2: per-row bitonic argsort + rank + z + x_c ----------------
__global__ void sort_row_kernel(const float* __restrict__ d,
                                const float* __restrict__ c,
                                int*   __restrict__ isort_out,
                                int*   __restrict__ k_out,
                                float* __restrict__ z_out,
                                float* __restrict__ xc_out) {
    __shared__ float skey[NU];
    __shared__ int   sidx[NU];
    __shared__ int   srank[NU];

    const int row = blockIdx.x;
    const int t   = threadIdx.x;            // 256 threads
    const float* drow = d + (size_t)row * NU;

    for (int u = t; u < NU; u += 256) {
        skey[u] = drow[u];
        sidx[u] = u;
    }
    __syncthreads();

    // Bitonic sort, ascending, 512 keys, 256 threads (one compare-exchange each).
    for (int k = 2; k <= NU; k <<= 1) {
        for (int j = k >> 1; j > 0; j >>= 1) {
            const int i1 = ((t & ~(j - 1)) << 1) | (t & (j - 1));
            const int i2 = i1 | j;
            const bool up = ((i1 & k) == 0);
            float k1 = skey[i1], k2 = skey[i2];
            if ((k1 > k2) == up) {
                int v1 = sidx[i1], v2 = sidx[i2];
                skey[i1] = k2; skey[i2] = k1;
                sidx[i1] = v2; sidx[i2] = v1;
            }
            __syncthreads();
        }
    }

    // ranks: srank[unit] = sorted position
    for (int p = t; p < NU; p += 256) srank[sidx[p]] = p;
    __syncthreads();

    for (int u = t; u < NU; u += 256) {
        isort_out[(size_t)row * NU + u] = sidx[u];
        const int r = srank[u];
        k_out[(size_t)row * NU + u] = r;
        z_out[(size_t)row * NU + u] = (r < NTOPK) ? (1.0f / (float)(r + 1)) : 0.0f;
    }

    // x_c = c[i_sort[0]]
    const int best = sidx[0];
    if (t < NF) xc_out[(size_t)row * NF + t] = c[(size_t)best * NF + t];
}

extern "C" void kernel_launch(void* const* d_in, const int* in_sizes, int n_in,
                              void* d_out, int out_size, void* d_ws, size_t ws_size,
                              hipStream_t stream) {
    const float* x = (const float*)d_in[0];
    const float* c = (const float*)d_in[1];
    // d_in[2] is topk (==8), compile-time constant here.

    float* out = (float*)d_out;
    float* d_dist   = out;                                       // 8192*512
    int*   d_isort  = (int*)(out + (size_t)NB * NU);             // 8192*512
    int*   d_k      = (int*)(out + 2 * (size_t)NB * NU);         // 8192*512
    float* d_z      = out + 3 * (size_t)NB * NU;                 // 8192*512
    float* d_xc     = out + 4 * (size_t)NB * NU;                 // 8192*64

    float* xnorm = (float*)d_ws;        // 8192 floats
    float* cnorm = xnorm + NB;          // 512 floats

    // Kernel 0: norms
    {
        int total = NB + NU;
        int blocks = (total + 255) / 256;
        norms_kernel<<<blocks, 256, 0, stream>>>(x, c, xnorm, cnorm);
    }

    // Kernel 1: distances via WMMA. 512*32 = 16384 tiles, 8 waves/block.
    {
        int tiles = (NB / 16) * (NU / 16);
        int wavesPerBlock = 8;
        int blocks = tiles / wavesPerBlock;
        dist_wmma_kernel<<<blocks, wavesPerBlock * 32, 0, stream>>>(x, c, xnorm, cnorm, d_dist);
    }

    // Kernel 2: per-row sort + outputs
    {
        sort_row_kernel<<<NB, 256, 0, stream>>>(d_dist, c, d_isort, d_k, d_z, d_xc);
    }
}